// PennyLaneQuantumCircuit_62835371541010
// MI455X (gfx1250) — compile-verified
//
#include <hip/hip_runtime.h>
#include <hip/hip_bf16.h>

typedef __attribute__((ext_vector_type(2))) float v2f;
typedef __attribute__((ext_vector_type(8))) float v8f;

#define NQ 8
#define NSTATES 256
#define NLAYERS 3
#define WAVES 2   // waves per block in the GEMM kernel

// ---------------------------------------------------------------------------
// Kernel 1: build the fixed 256x256 circuit unitary U (the 3 variational
// layers; the per-batch RY encoding is handled in kernel 2) and store it in
// WMMA-A-fragment swizzled layout so the GEMM reads it with coalesced b128:
//   element U[s][t], s = mtile*16 + mrow, t = 4k + kk:
//     lane = mrow + 16*(kk>=2), v = kk&1
//     Uswz[((mtile*64 + k)*32 + lane)*4 + v]     = Re
//     Uswz[((mtile*64 + k)*32 + lane)*4 + 2 + v] = Im
// (matches ISA 32-bit A-matrix 16x4 layout: lanes 0-15 hold K=0,1 in v0,v1;
//  lanes 16-31 hold K=2,3)
// ---------------------------------------------------------------------------
__global__ void build_unitary(const float* __restrict__ theta,
                              const float* __restrict__ omega,
                              float* __restrict__ Uswz) {
  __shared__ float2 st[NSTATES];
  const int s = threadIdx.x;   // amplitude (row) index
  const int t = blockIdx.x;    // basis column index
  st[s] = make_float2(s == t ? 1.0f : 0.0f, 0.0f);
  __syncthreads();

  for (int l = 0; l < NLAYERS; ++l) {
    // RY(theta[l][q]) on wire q  (wire q <-> bit 7-q of the flat index)
    for (int q = 0; q < NQ; ++q) {
      const int p = 7 - q;
      const float h = 0.5f * theta[l * NQ + q];
      const float c = cosf(h), sn = sinf(h);
      if (((s >> p) & 1) == 0) {
        const int pr = s | (1 << p);
        float2 a0 = st[s], a1 = st[pr];
        st[s]  = make_float2(c * a0.x - sn * a1.x, c * a0.y - sn * a1.y);
        st[pr] = make_float2(sn * a0.x + c * a1.x, sn * a0.y + c * a1.y);
      }
      __syncthreads();
    }
    // CNOT ring: ctrl wire q, tgt wire (q+1)%8, applied sequentially
    for (int q = 0; q < NQ; ++q) {
      const int pc = 7 - q;
      const int pt = 7 - ((q + 1) & 7);
      if ((((s >> pc) & 1) == 1) && (((s >> pt) & 1) == 0)) {
        const int pr = s | (1 << pt);
        float2 tmp = st[s]; st[s] = st[pr]; st[pr] = tmp;
      }
      __syncthreads();
    }
    // RZ(omega[l][q]) are all diagonal: accumulate one total phase per s
    float ang = 0.0f;
    for (int q = 0; q < NQ; ++q) {
      const int p = 7 - q;
      const float h = 0.5f * omega[l * NQ + q];
      ang += ((s >> p) & 1) ? h : -h;
    }
    {
      const float c = cosf(ang), sn = sinf(ang);
      float2 a = st[s];
      st[s] = make_float2(a.x * c - a.y * sn, a.x * sn + a.y * c);
    }
    __syncthreads();
  }

  // swizzled fragment store
  const int mtile = s >> 4, mrow = s & 15;
  const int k = t >> 2, kk = t & 3;
  const int lane = mrow + ((kk & 2) ? 16 : 0);
  const int v = kk & 1;
  const int idx = ((mtile * 64 + k) * 32 + lane) * 4;
  float2 a = st[s];
  Uswz[idx + v]     = a.x;
  Uswz[idx + 2 + v] = a.y;
}

// ---------------------------------------------------------------------------
// Kernel 2: fused  Final = U @ Init  (complex x real, fp32 WMMA 16x16x4)
// plus |.|^2 and the signed (Walsh) reduction to <Z_q>.
// One wave owns 16 batch columns; loops all 16 M-tiles so the M-reduction
// stays in registers. B fragments (Init) are built once into LDS in exact
// B-matrix fragment order:  entry (k*32+lane)*2 + v  =  Init[4k + 2*laneHi + v][lane&15].
// ---------------------------------------------------------------------------
__global__ __launch_bounds__(WAVES * 32) void qexp_gemm(
    const float* __restrict__ x, const float* __restrict__ Uswz,
    float* __restrict__ out) {
  __shared__ float ldsInit[WAVES * 64 * 32 * 2];   // 16 KB per wave
  const int lane   = threadIdx.x & 31;
  const int wave   = threadIdx.x >> 5;
  const int ntile  = blockIdx.x * WAVES + wave;
  const int n      = lane & 15;
  const int laneHi = lane >> 4;
  const int b      = ntile * 16 + n;

  // encoding factors for this lane's batch element
  float c[NQ], sn[NQ];
  #pragma unroll
  for (int q = 0; q < NQ; ++q) {
    const float ang = 0.5f * x[b * NQ + q];
    c[q] = cosf(ang); sn[q] = sinf(ang);
  }

  // Init[t][n] = prod_q (bit_{7-q}(t) ? sin : cos);  t = 4k + 2*laneHi + v
  // t bit1 = laneHi (wire 6), t bit0 = v (wire 7), t bits 2..7 = k bits 0..5
  // (wires 5..0)
  float* myLds = &ldsInit[wave * (64 * 32 * 2)];
  const float f6 = laneHi ? sn[6] : c[6];
  for (int k = 0; k < 64; ++k) {
    float ph = (k & 1)  ? sn[5] : c[5];
    ph      *= (k & 2)  ? sn[4] : c[4];
    ph      *= (k & 4)  ? sn[3] : c[3];
    ph      *= (k & 8)  ? sn[2] : c[2];
    ph      *= (k & 16) ? sn[1] : c[1];
    ph      *= (k & 32) ? sn[0] : c[0];
    const float g = ph * f6;
    const int o = (k * 32 + lane) * 2;
    myLds[o]     = g * c[7];    // v = 0
    myLds[o + 1] = g * sn[7];   // v = 1
  }
  __syncthreads();

  float accq[NQ];
  #pragma unroll
  for (int q = 0; q < NQ; ++q) accq[q] = 0.0f;

  const float4* __restrict__ Ufrag = (const float4*)Uswz;

  for (int mt = 0; mt < 16; ++mt) {
    v8f ar = {0.f, 0.f, 0.f, 0.f, 0.f, 0.f, 0.f, 0.f};
    v8f ai = {0.f, 0.f, 0.f, 0.f, 0.f, 0.f, 0.f, 0.f};
    #pragma unroll 8
    for (int k = 0; k < 64; ++k) {
      const float4 f = Ufrag[(mt * 64 + k) * 32 + lane];   // coalesced b128
      v2f a_re = {f.x, f.y};
      v2f a_im = {f.z, f.w};
      const float2 bb = *(const float2*)&myLds[(k * 32 + lane) * 2];
      v2f bf = {bb.x, bb.y};
      // D = A(16x4) x B(4x16) + C, fp32 exact
      ar = __builtin_amdgcn_wmma_f32_16x16x4_f32(
          false, a_re, false, bf, (short)0, ar, false, false);
      ai = __builtin_amdgcn_wmma_f32_16x16x4_f32(
          false, a_im, false, bf, (short)0, ai, false, false);
    }
    // probabilities for this 16x16 tile (lane holds m = v + 8*laneHi, n)
    float p[8];
    #pragma unroll
    for (int v = 0; v < 8; ++v) p[v] = ar[v] * ar[v] + ai[v] * ai[v];

    // Walsh sums over v (s-bits 2..0 -> qubits 5,6,7)
    const float s01 = p[0] + p[1], s23 = p[2] + p[3];
    const float s45 = p[4] + p[5], s67 = p[6] + p[7];
    const float S   = (s01 + s23) + (s45 + s67);
    const float Sb2 = (s01 + s23) - (s45 + s67);                       // q=5
    const float Sb1 = (s01 - s23) + (s45 - s67);                       // q=6
    const float Sb0 = (p[0]-p[1]) + (p[2]-p[3]) + (p[4]-p[5]) + (p[6]-p[7]); // q=7
    accq[5] += Sb2; accq[6] += Sb1; accq[7] += Sb0;
    accq[4] += laneHi   ? -S : S;   // s-bit3 = laneHi
    accq[0] += (mt & 8) ? -S : S;   // s-bit7
    accq[1] += (mt & 4) ? -S : S;   // s-bit6
    accq[2] += (mt & 2) ? -S : S;   // s-bit5
    accq[3] += (mt & 1) ? -S : S;   // s-bit4
  }

  // combine lane halves (m=0..7 with m=8..15 share the same column n)
  #pragma unroll
  for (int q = 0; q < NQ; ++q) accq[q] += __shfl_xor(accq[q], 16, 32);

  if (lane < 16) {
    #pragma unroll
    for (int q = 0; q < NQ; ++q) out[b * NQ + q] = accq[q];
  }
}

// ---------------------------------------------------------------------------
extern "C" void kernel_launch(void* const* d_in, const int* in_sizes, int n_in,
                              void* d_out, int out_size, void* d_ws, size_t ws_size,
                              hipStream_t stream) {
  const float* x     = (const float*)d_in[0];   // (B, 8)
  const float* theta = (const float*)d_in[1];   // (3, 8)
  const float* omega = (const float*)d_in[2];   // (3, 8)
  float* out  = (float*)d_out;                  // (B, 8)
  float* Uswz = (float*)d_ws;                   // 256*256*2 floats = 512 KB

  const int B = in_sizes[0] / NQ;
  const int ntiles = B / 16;

  build_unitary<<<NSTATES, NSTATES, 0, stream>>>(theta, omega, Uswz);
  qexp_gemm<<<ntiles / WAVES, WAVES * 32, 0, stream>>>(x, Uswz, out);
}